// Zoom_89026082111878
// MI455X (gfx1250) — compile-verified
//
#include <hip/hip_runtime.h>
#include <hip/hip_bf16.h>

// Zoom (frequency-axis linear interpolation) for x:[B,C,T,F]=[8,16,256,1024] f32, k=2, n=3.
// Bandwidth problem: read only the touched prefix (~86 MB) + write 128 MB ~ 9.2 us @ 23.3 TB/s.
// CDNA5 paths: TDM tensor_load_to_lds stages a [nrows x nload] f32 tile per WG into LDS,
// s_wait_tensorcnt + barrier, LDS gather, non-temporal float4 stores.
// The gather coefficients depend only on f (same for all 32768 rows), so a tiny one-block
// kernel precomputes a 16KB (w0,w1,i0,i1) table + nload into d_ws; the main kernel does
// zero integer/float divides.

#define F_DIM       1024
#define ROWS_PER_WG 16
#define TBL_BYTES   (F_DIM * 16)   // 1024 float4 entries

typedef unsigned int v4u __attribute__((ext_vector_type(4)));
typedef int          v4i __attribute__((ext_vector_type(4)));
typedef int          v8i __attribute__((ext_vector_type(8)));
typedef float        f4  __attribute__((ext_vector_type(4)));

// ---------------- Kernel A: coefficient table (1 block, 256 threads) ----------------
__global__ __launch_bounds__(256) void zoom_build_table(
    const int* __restrict__ kp, const int* __restrict__ np,
    f4* __restrict__ tbl, int* __restrict__ nload_out)
{
    const int k = *kp;
    const int n = *np;
    const int t = (int)threadIdx.x;

    if (t == 0) {
        const long long pmax = (long long)(F_DIM - 1) * (long long)k;
        int nload = (int)(pmax / (long long)n) + 2;
        if (nload > F_DIM) nload = F_DIM;
        *nload_out = nload;
    }

#pragma unroll
    for (int j = 0; j < 4; ++j) {
        const int f   = t * 4 + j;
        const int pos = f * k;
        const int idx = pos / n;
        const int rem = pos - idx * n;
        const float frac = (float)rem / (float)n;  // matches reference's f32 divide
        const float w    = 1.0f - frac;
        const bool v0 = idx < F_DIM;
        const bool v1 = (idx + 1) < F_DIM;
        f4 e;
        e.x = v0 ? w : 0.0f;
        e.y = v1 ? (1.0f - w) : 0.0f;
        e.z = __int_as_float(v0 ? idx : 0);
        e.w = __int_as_float(v1 ? (idx + 1) : 0);
        tbl[f] = e;
    }
}

// ---------------- Kernel B: TDM-staged interpolation ----------------
template <bool USE_TABLE>
__global__ __launch_bounds__(256) void zoom_tdm_kernel(
    const float* __restrict__ x,
    const f4*    __restrict__ tbl,      // USE_TABLE only
    const int*   __restrict__ nlp,      // USE_TABLE only: precomputed nload
    const int*   __restrict__ kp,       // fallback only
    const int*   __restrict__ np,       // fallback only
    float*       __restrict__ out,
    int rows)
{
    __shared__ float tile[ROWS_PER_WG * F_DIM];   // 64 KB

    int nload;
    if constexpr (USE_TABLE) {
        nload = __builtin_amdgcn_readfirstlane(*nlp);
    } else {
        const int k = __builtin_amdgcn_readfirstlane(*kp);
        const int n = __builtin_amdgcn_readfirstlane(*np);
        const long long pmax = (long long)(F_DIM - 1) * (long long)k;
        nload = (int)(pmax / (long long)n) + 2;
        if (nload > F_DIM) nload = F_DIM;
    }

    const int row0 = blockIdx.x * ROWS_PER_WG;
    int nrows = rows - row0;
    if (nrows <= 0) return;                        // uniform per block
    if (nrows > ROWS_PER_WG) nrows = ROWS_PER_WG;

    // ---- TDM: DMA the [nrows x nload] f32 tile (row stride F_DIM) into LDS ----
    // TDM ignores EXEC; issue from exactly one wave via a scalar (readfirstlane) branch.
    if (__builtin_amdgcn_readfirstlane((int)threadIdx.x) < 32) {
        const unsigned long long ga =
            (unsigned long long)(const void*)(x + (size_t)row0 * F_DIM);
        const unsigned ldsoff = (unsigned)(unsigned long long)(void*)&tile[0];
        const unsigned dim1   = (unsigned)(rows - row0);  // remaining tensor rows (OOB guard)

        // D# group 0: count=1 | lds_addr | global_addr[56:0] | type=2
        v4u g0 = { 1u,
                   ldsoff,
                   (unsigned)(ga & 0xFFFFFFFFull),
                   ((unsigned)((ga >> 32) & 0x01FFFFFFull)) | 0x80000000u };

        // D# group 1: data_size=4B; tensor_dim0=F; tensor_dim1=dim1; tile_dim0=nload;
        //             tile_dim1=nrows; tile_dim2=0; tensor_dim0_stride=F.
        v8i g1 = { (int)0x00020000,
                   (int)(((unsigned)F_DIM & 0xFFFFu) << 16),
                   (int)((((unsigned)F_DIM >> 16) & 0xFFFFu) | ((dim1 & 0xFFFFu) << 16)),
                   (int)(((dim1 >> 16) & 0xFFFFu) | (((unsigned)nload & 0xFFFFu) << 16)),
                   (int)((unsigned)nrows & 0xFFFFu),
                   (int)F_DIM,
                   0, 0 };
        v4i gz4 = { 0, 0, 0, 0 };                 // groups 2/3: 2-D tile, unused
        v8i gz8 = { 0, 0, 0, 0, 0, 0, 0, 0 };

        __builtin_amdgcn_tensor_load_to_lds(g0, g1, gz4, gz4, gz8, 0);
        __builtin_amdgcn_s_wait_tensorcnt(0);
    }

    // ---- Per-thread gather coefficients for 4 consecutive outputs (reused per row) ----
    const int t  = (int)threadIdx.x;
    const int f0 = t << 2;                         // 256 threads * 4 = 1024 = F_DIM
    float w0[4], w1[4];
    int   i0[4], i1[4];
    if constexpr (USE_TABLE) {
#pragma unroll
        for (int j = 0; j < 4; ++j) {
            const f4 e = tbl[f0 + j];              // 16KB table, L2-hot
            w0[j] = e.x;
            w1[j] = e.y;
            i0[j] = __float_as_int(e.z);
            i1[j] = __float_as_int(e.w);
        }
    } else {
        const int k = __builtin_amdgcn_readfirstlane(*kp);
        const int n = __builtin_amdgcn_readfirstlane(*np);
#pragma unroll
        for (int j = 0; j < 4; ++j) {
            const int f   = f0 + j;
            const int pos = f * k;
            const int idx = pos / n;
            const int rem = pos - idx * n;
            const float frac = (float)rem / (float)n;
            const float w    = 1.0f - frac;
            const bool v0 = idx < F_DIM;
            const bool v1 = (idx + 1) < F_DIM;
            w0[j] = v0 ? w : 0.0f;
            w1[j] = v1 ? (1.0f - w) : 0.0f;
            i0[j] = v0 ? idx : 0;                  // nonzero weights imply idx < nload
            i1[j] = v1 ? (idx + 1) : 0;
        }
    }

    __syncthreads();                               // TDM tile visible to all waves

    // ---- Interpolate each row out of LDS, stream results with NT float4 stores ----
    for (int r = 0; r < nrows; ++r) {
        const float* __restrict__ ls = &tile[r * nload];
        f4 o;
#pragma unroll
        for (int j = 0; j < 4; ++j)
            o[j] = w0[j] * ls[i0[j]] + w1[j] * ls[i1[j]];
        f4* dst = (f4*)(out + (size_t)(row0 + r) * F_DIM + f0);
        __builtin_nontemporal_store(o, dst);
    }
}

extern "C" void kernel_launch(void* const* d_in, const int* in_sizes, int n_in,
                              void* d_out, int out_size, void* d_ws, size_t ws_size,
                              hipStream_t stream) {
    const float* x  = (const float*)d_in[0];
    const int*   kp = (const int*)d_in[1];
    const int*   np = (const int*)d_in[2];
    float*       o  = (float*)d_out;

    const int rows = in_sizes[0] / F_DIM;          // 8*16*256 = 32768
    const int grid = (rows + ROWS_PER_WG - 1) / ROWS_PER_WG;

    if (ws_size >= (size_t)(TBL_BYTES + 16)) {
        f4*  tbl = (f4*)d_ws;
        int* nlp = (int*)((char*)d_ws + TBL_BYTES);
        zoom_build_table<<<1, 256, 0, stream>>>(kp, np, tbl, nlp);
        zoom_tdm_kernel<true><<<grid, 256, 0, stream>>>(x, tbl, nlp, kp, np, o, rows);
    } else {
        zoom_tdm_kernel<false><<<grid, 256, 0, stream>>>(x, nullptr, nullptr, kp, np, o, rows);
    }
}